// HypergraphAttnConv_11922829213806
// MI455X (gfx1250) — compile-verified
//
#include <hip/hip_runtime.h>

#define N_NODES 100000
#define N_EDGES 1600000
#define IN_CH   128
#define OUT_CH  64
#define NEG_SLOPE 0.2f
#define EPS_F   1e-16f

typedef __attribute__((ext_vector_type(2))) float v2f;
typedef __attribute__((ext_vector_type(8))) float v8f;

// ---------------------------------------------------------------------------
// Kernel 0: grid-stride zero fill (out and denom must start at 0 every call)
// ---------------------------------------------------------------------------
__global__ void gat_zero_kernel(float* __restrict__ p, long n) {
    long i = (long)blockIdx.x * blockDim.x + threadIdx.x;
    long stride = (long)gridDim.x * blockDim.x;
    for (; i < n; i += stride) p[i] = 0.0f;
}

// ---------------------------------------------------------------------------
// Kernel 1: h = x @ W^T via V_WMMA_F32_16X16X4_F32, fused s_src/s_dst.
// Block = 128 threads (4 wave32). Block b computes rows [16b, 16b+16).
// Wave w computes output-channel tile [16w, 16w+16). W staged in LDS.
// ---------------------------------------------------------------------------
__global__ __launch_bounds__(128)
void gat_gemm_kernel(const float* __restrict__ x, const float* __restrict__ W,
                     const float* __restrict__ att,
                     float* __restrict__ h,
                     float* __restrict__ s_src, float* __restrict__ s_dst)
{
    __shared__ float Wlds[OUT_CH * 132];   // 64 x 128, padded stride 132 (bank spread)
    __shared__ float sred[2][16];          // per-row partial dot with att_src/att_dst

    const int tid = threadIdx.x;

    // Cooperative copy of W (64x128 fp32, row-major) into padded LDS.
    for (int i = tid; i < (OUT_CH * IN_CH) / 4; i += 128) {
        const int o = (i * 4) / IN_CH;
        const int k = (i * 4) % IN_CH;
        const float4 wv = *(const float4*)(W + o * IN_CH + k);
        float* dst = &Wlds[o * 132 + k];
        dst[0] = wv.x; dst[1] = wv.y; dst[2] = wv.z; dst[3] = wv.w;
    }
    if (tid < 32) ((float*)sred)[tid] = 0.0f;
    __syncthreads();

    const int lane = tid & 31;
    const int wave = tid >> 5;
    const int mn   = lane & 15;           // A-row within tile / B-col within tile
    const int kg   = (lane >> 4) * 2;     // K sub-offset inside each 4-wide slab
    const int rowBase = blockIdx.x * 16;
    const int n = wave * 16 + mn;         // this lane's output channel

    const float* aptr = x + (size_t)(rowBase + mn) * IN_CH + kg;
    const float* bptr = &Wlds[n * 132 + kg];

    v8f c = {};
    #pragma unroll
    for (int k0 = 0; k0 < IN_CH; k0 += 4) {
        v2f a; a.x = aptr[k0]; a.y = aptr[k0 + 1];   // A 16x4 fragment
        v2f b; b.x = bptr[k0]; b.y = bptr[k0 + 1];   // B 4x16 fragment (from LDS)
        c = __builtin_amdgcn_wmma_f32_16x16x4_f32(false, a, false, b,
                                                  (short)0, c, false, false);
    }

    // C/D layout: lane, VGPR v -> M = v + 8*(lane>>4), N = lane&15.
    const float as = att[n];
    const float ad = att[OUT_CH + n];
    #pragma unroll
    for (int v = 0; v < 8; ++v) {
        const int mr = v + 8 * (lane >> 4);
        h[(size_t)(rowBase + mr) * OUT_CH + n] = c[v];
        atomicAdd(&sred[0][mr], c[v] * as);   // ds_add_f32
        atomicAdd(&sred[1][mr], c[v] * ad);
    }
    __syncthreads();
    if (tid < 16) {
        s_src[rowBase + tid] = sred[0][tid];
        s_dst[rowBase + tid] = sred[1][tid];
    }
}

// ---------------------------------------------------------------------------
// Kernel 2: per-edge unnormalized attention + segment-sum denominator.
// ---------------------------------------------------------------------------
__global__ __launch_bounds__(256)
void gat_edge_score_kernel(const long long* __restrict__ ei,
                           const float* __restrict__ s_src,
                           const float* __restrict__ s_dst,
                           float* __restrict__ alpha,
                           float* __restrict__ denom)
{
    const int e = blockIdx.x * 256 + threadIdx.x;
    if (e >= N_EDGES) return;
    const int row = (int)ei[e];
    const int col = (int)ei[N_EDGES + e];
    float v = s_src[row] + s_dst[col];
    v = v > 0.0f ? v : NEG_SLOPE * v;        // leaky_relu
    const float a = __expf(v);
    alpha[e] = a;
    atomicAdd(&denom[col], a);
}

// ---------------------------------------------------------------------------
// Kernel 3: out[col] += (alpha/denom[col]) * h[row].  64 threads per edge so
// h-reads and out-atomics are contiguous 256B segments.
// ---------------------------------------------------------------------------
__global__ __launch_bounds__(256)
void gat_aggregate_kernel(const long long* __restrict__ ei,
                          const float* __restrict__ h,
                          const float* __restrict__ alpha,
                          const float* __restrict__ denom,
                          float* __restrict__ out)
{
    const long t = (long)blockIdx.x * 256 + threadIdx.x;
    const int e  = (int)(t >> 6);
    const int ch = (int)(t & 63);
    if (e >= N_EDGES) return;
    const int row = (int)ei[e];
    const int col = (int)ei[N_EDGES + e];
    const float a = alpha[e] / (denom[col] + EPS_F);
    atomicAdd(&out[(size_t)col * OUT_CH + ch], a * h[(size_t)row * OUT_CH + ch]);
}

// ---------------------------------------------------------------------------
extern "C" void kernel_launch(void* const* d_in, const int* in_sizes, int n_in,
                              void* d_out, int out_size, void* d_ws, size_t ws_size,
                              hipStream_t stream)
{
    const float*     x   = (const float*)d_in[0];     // (100000, 128) fp32
    const float*     W   = (const float*)d_in[1];     // (64, 128)     fp32
    const float*     att = (const float*)d_in[2];     // (1, 128)      fp32
    const long long* ei  = (const long long*)d_in[3]; // (2, 1600000)  int64
    float* out = (float*)d_out;                       // (100000, 64)  fp32

    float* ws    = (float*)d_ws;
    float* h     = ws;                                     // 6,400,000
    float* s_src = h + (size_t)N_NODES * OUT_CH;           // 100,000
    float* s_dst = s_src + N_NODES;                        // 100,000
    float* denom = s_dst + N_NODES;                        // 100,000
    float* alpha = denom + N_NODES;                        // 1,600,000

    // Zero the accumulation targets.
    gat_zero_kernel<<<2048, 256, 0, stream>>>(out, (long)N_NODES * OUT_CH);
    gat_zero_kernel<<<512,  256, 0, stream>>>(denom, (long)N_NODES);

    // Phase 1: WMMA GEMM + fused attention projections.
    gat_gemm_kernel<<<N_NODES / 16, 128, 0, stream>>>(x, W, att, h, s_src, s_dst);

    // Phase 2: edge scores + softmax denominator.
    gat_edge_score_kernel<<<(N_EDGES + 255) / 256, 256, 0, stream>>>(
        ei, s_src, s_dst, alpha, denom);

    // Phase 3: normalized weighted aggregation.
    const long work = (long)N_EDGES * OUT_CH;
    gat_aggregate_kernel<<<(unsigned)((work + 255) / 256), 256, 0, stream>>>(
        ei, h, alpha, denom, out);
}